// Based_18399639896655
// MI455X (gfx1250) — compile-verified
//
#include <hip/hip_runtime.h>

typedef float v2f __attribute__((ext_vector_type(2)));
typedef float v8f __attribute__((ext_vector_type(8)));

#define WMMA_F32(a, b, c) \
  __builtin_amdgcn_wmma_f32_16x16x4_f32(false, (a), false, (b), (short)0, (c), false, false)

constexpr int L = 1024, DM = 768, Hh = 12, FD = 16, HD = 64;
constexpr float EPSF = 1e-12f;
constexpr float QSC = 0.17677669529663687f; // 1/(4*sqrt(2)) quad feature scale

// ---------------------------------------------------------------------------
// f32 GEMM: C[M,N] = A[M,K] @ B[K,N] via V_WMMA_F32_16X16X4_F32.
// Block = 256 threads (8 wave32). Tile 64x64, K-step 32. Each wave owns a
// 16x32 strip (two 16x16 subtiles). float4 global staging + register
// prefetch. B is stored TRANSPOSED in LDS so each lane's (k,k+1) fragment
// pair is contiguous -> single 64-bit LDS loads, no register shuffles.
// All problem sizes are multiples of (64,64,32): no edge guards.
// ---------------------------------------------------------------------------
__global__ __launch_bounds__(256) void gemm_f32_wmma(const float* __restrict__ A,
                                                     const float* __restrict__ B,
                                                     float* __restrict__ C,
                                                     int M, int N, int K) {
  __shared__ float As[64][36];    // 64 rows x 32 k (+4 pad)
  __shared__ float BsT[64][38];   // 64 cols x 32 k (+6 pad, even stride)
  const int tid  = threadIdx.x;
  const int wave = tid >> 5, lane = tid & 31;
  const int kh   = lane >> 4, ln = lane & 15;
  const int ntb  = N / 64;
  const int bm   = (blockIdx.x / ntb) * 64;
  const int bn   = (blockIdx.x % ntb) * 64;
  const int wm   = wave >> 1;    // 0..3  -> row subtile
  const int wn   = wave & 1;     // 0..1  -> 32-col strip
  const int ar   = tid >> 2, ac = (tid & 3) * 4;   // A stage: row 0..63, col {0,4,8,12}
  const int br   = tid >> 3, bc = (tid & 7) * 4;   // B stage: row 0..31, col {0..28}
  (void)M;

  const float4* Ag = (const float4*)A;
  const float4* Bg = (const float4*)B;

  float4 a0 = Ag[((size_t)(bm + ar) * K + ac) >> 2];
  float4 a1 = Ag[((size_t)(bm + ar) * K + ac + 16) >> 2];
  float4 b0 = Bg[((size_t)br * N + bn + bc) >> 2];
  float4 b1 = Bg[((size_t)br * N + bn + bc + 32) >> 2];
  v8f acc0 = {}, acc1 = {};

  for (int kb = 0; kb < K; kb += 32) {
    *(float4*)&As[ar][ac]      = a0;
    *(float4*)&As[ar][ac + 16] = a1;
    BsT[bc + 0][br]  = b0.x;  BsT[bc + 1][br]  = b0.y;
    BsT[bc + 2][br]  = b0.z;  BsT[bc + 3][br]  = b0.w;
    BsT[bc + 32][br] = b1.x;  BsT[bc + 33][br] = b1.y;
    BsT[bc + 34][br] = b1.z;  BsT[bc + 35][br] = b1.w;
    __syncthreads();
    if (kb + 32 < K) {  // prefetch next stage; overlaps WMMA below
      a0 = Ag[((size_t)(bm + ar) * K + kb + 32 + ac) >> 2];
      a1 = Ag[((size_t)(bm + ar) * K + kb + 32 + ac + 16) >> 2];
      b0 = Bg[((size_t)(kb + 32 + br) * N + bn + bc) >> 2];
      b1 = Bg[((size_t)(kb + 32 + br) * N + bn + bc + 32) >> 2];
    }
#pragma unroll
    for (int k = 0; k < 32; k += 4) {
      const int kk = k + 2 * kh;
      v2f a = *(const v2f*)&As[wm * 16 + ln][kk];
      v2f p = *(const v2f*)&BsT[wn * 32 + ln][kk];
      v2f q = *(const v2f*)&BsT[wn * 32 + 16 + ln][kk];
      acc0 = WMMA_F32(a, p, acc0);
      acc1 = WMMA_F32(a, q, acc1);
    }
    __syncthreads();
  }
#pragma unroll
  for (int r = 0; r < 8; ++r) {
    const size_t row = (size_t)(bm + wm * 16 + r + 8 * kh);
    C[row * N + bn + wn * 32 + ln]      = acc0[r];
    C[row * N + bn + wn * 32 + 16 + ln] = acc1[r];
  }
}

// ---------------------------------------------------------------------------
// Chunked causal linear attention with Taylor feature map.
// Grid: 48 blocks = head (12) x v-block (4 x 16 cols). 256 threads / block.
// phi(q).phi(k) = 1 + s/4 + s^2/32 with s = q.k   (intra-chunk, no features)
// Inter-chunk state split: ones row (16) + linear (16x16) + quad (256x16).
// ---------------------------------------------------------------------------
__global__ __launch_bounds__(256) void based_attn(const float* __restrict__ Qg,
                                                  const float* __restrict__ Kg,
                                                  const float* __restrict__ Vg,
                                                  float* __restrict__ Yg) {
  __shared__ float s_lin[16][17];    // sum (k/2) v^T   (v-slice)
  __shared__ float s_quad[256][17];  // sum kq^T v      (v-slice)
  __shared__ float s_ones[16];       // sum v
  __shared__ float kcum[273];        // sum phi(k) (full, for denominator)
  __shared__ float Qc[32][20];
  __shared__ float Kc[32][20];
  __shared__ float Vc[32][20];
  __shared__ float Aw[32][33];       // masked polynomial attention weights
  __shared__ float Fq[32][257];      // Qq / Kq quadratic features (reused)
  __shared__ float Yc[32][20];
  __shared__ float den[32];

  const int tid  = threadIdx.x;
  const int wave = tid >> 5, lane = tid & 31;
  const int kh   = lane >> 4, ln = lane & 15;
  const int h    = blockIdx.x >> 2;
  const int vb   = blockIdx.x & 3;
  const int qoff = h * FD;
  const int voff = h * HD + vb * 16;

  for (int i = tid; i < 16 * 16; i += 256) s_lin[i >> 4][i & 15] = 0.f;
  for (int i = tid; i < 256 * 16; i += 256) s_quad[i >> 4][i & 15] = 0.f;
  if (tid < 16) s_ones[tid] = 0.f;
  for (int i = tid; i < 273; i += 256) kcum[i] = 0.f;
  __syncthreads();

  for (int c = 0; c < L / 32; ++c) {
    const int bl = c * 32;
    // ---- load chunk (float4 vectorized) ----
    if (tid < 128) {
      const int t = tid >> 2, c4 = (tid & 3) * 4;
      *(float4*)&Qc[t][c4] = *(const float4*)&Qg[(size_t)(bl + t) * (Hh * FD) + qoff + c4];
      *(float4*)&Kc[t][c4] = *(const float4*)&Kg[(size_t)(bl + t) * (Hh * FD) + qoff + c4];
    } else {
      const int i = tid - 128;
      const int t = i >> 2, c4 = (i & 3) * 4;
      *(float4*)&Vc[t][c4] = *(const float4*)&Vg[(size_t)(bl + t) * DM + voff + c4];
    }
    __syncthreads();  // (A)

    // ---- S = Qc Kc^T -> polynomial + causal mask (waves 0..3) ----
    if (wave < 4) {
      const int ti = wave >> 1, tj = wave & 1;
      v8f acc = {};
#pragma unroll
      for (int k = 0; k < 16; k += 4) {
        const int kk = k + 2 * kh;
        v2f a = *(const v2f*)&Qc[ti * 16 + ln][kk];
        v2f b = *(const v2f*)&Kc[tj * 16 + ln][kk];
        acc = WMMA_F32(a, b, acc);
      }
#pragma unroll
      for (int r = 0; r < 8; ++r) {
        const int row = ti * 16 + r + 8 * kh, col = tj * 16 + ln;
        const float s = acc[r];
        Aw[row][col] = (row >= col) ? (1.f + 0.25f * s + (s * s) * 0.03125f) : 0.f;
      }
    }
    // ---- quadratic features of Q -> Fq (all threads) ----
    for (int i = tid; i < 32 * 256; i += 256) {
      int t = i >> 8, ij = i & 255;
      Fq[t][ij] = Qc[t][ij >> 4] * Qc[t][ij & 15] * QSC;
    }
    __syncthreads();  // (B)

    // ---- Y = ones + Aw@Vc + (Qc/2)@s_lin + Fq@s_quad (waves 0,1) ----
    if (wave < 2) {
      const int ti = wave;
      const int row = ti * 16 + ln;
      v8f acc;
#pragma unroll
      for (int r = 0; r < 8; ++r) acc[r] = s_ones[ln];
#pragma unroll
      for (int k = 0; k < 32; k += 4) {
        const int kk = k + 2 * kh;
        v2f a = *(const v2f*)&Aw[row][kk];
        v2f b;
        b.x = Vc[kk][ln];  b.y = Vc[kk + 1][ln];
        acc = WMMA_F32(a, b, acc);
      }
#pragma unroll
      for (int k = 0; k < 16; k += 4) {
        const int kk = k + 2 * kh;
        v2f a, b;
        a.x = Qc[row][kk] * 0.5f; a.y = Qc[row][kk + 1] * 0.5f;
        b.x = s_lin[kk][ln];      b.y = s_lin[kk + 1][ln];
        acc = WMMA_F32(a, b, acc);
      }
#pragma unroll 8
      for (int k = 0; k < 256; k += 4) {
        const int kk = k + 2 * kh;
        v2f a = *(const v2f*)&Fq[row][kk];
        v2f b;
        b.x = s_quad[kk][ln]; b.y = s_quad[kk + 1][ln];
        acc = WMMA_F32(a, b, acc);
      }
#pragma unroll
      for (int r = 0; r < 8; ++r) Yc[ti * 16 + r + 8 * kh][ln] = acc[r];
    } else if (wave == 2) {
      // ---- denominator: rowsum(Aw) + phi(q).kcum ----
      const int t = lane;
      float d = kcum[0];
      for (int s2 = 0; s2 < 32; ++s2) d += Aw[t][s2];
      for (int i = 0; i < 16; ++i) d += Qc[t][i] * 0.5f * kcum[1 + i];
      for (int ij = 0; ij < 256; ++ij) d += Fq[t][ij] * kcum[17 + ij];
      den[t] = d + EPSF;
    }
    __syncthreads();  // (C)

    // ---- write normalized output slice (float4) ----
    if (tid < 128) {
      const int t = tid >> 2, c4 = (tid & 3) * 4;
      const float dd = den[t];
      float4 y = *(float4*)&Yc[t][c4];
      y.x /= dd; y.y /= dd; y.z /= dd; y.w /= dd;
      *(float4*)&Yg[(size_t)(bl + t) * DM + voff + c4] = y;
    }
    // ---- quadratic features of K (overwrite Fq) ----
    for (int i = tid; i < 32 * 256; i += 256) {
      int t = i >> 8, ij = i & 255;
      Fq[t][ij] = Kc[t][ij >> 4] * Kc[t][ij & 15] * QSC;
    }
    // ---- scalar state pieces ----
    if (tid >= 128 && tid < 144) {
      const int f = tid - 128;
      float s = 0.f;
      for (int t = 0; t < 32; ++t) s += Vc[t][f];
      s_ones[f] += s;
    } else if (tid >= 144 && tid < 160) {
      const int f = tid - 144;
      float s = 0.f;
      for (int t = 0; t < 32; ++t) s += Kc[t][f];
      kcum[1 + f] += 0.5f * s;
    } else if (tid == 160) {
      kcum[0] += 32.f;
    }
    __syncthreads();  // (D)

    // ---- kcum quadratic part ----
    {
      float s = 0.f;
      for (int t = 0; t < 32; ++t) s += Fq[t][tid];
      kcum[17 + tid] += s;
    }
    // ---- state GEMM updates: tiles 0..15 quad, tile 16 lin ----
    for (int tile = wave; tile < 17; tile += 8) {
      v8f acc;
      if (tile < 16) {
        const int qm = tile * 16;
#pragma unroll
        for (int r = 0; r < 8; ++r) acc[r] = s_quad[qm + r + 8 * kh][ln];
#pragma unroll
        for (int k = 0; k < 32; k += 4) {
          const int kk = k + 2 * kh;
          v2f a, b;
          a.x = Fq[kk][qm + ln]; a.y = Fq[kk + 1][qm + ln];
          b.x = Vc[kk][ln];      b.y = Vc[kk + 1][ln];
          acc = WMMA_F32(a, b, acc);
        }
#pragma unroll
        for (int r = 0; r < 8; ++r) s_quad[qm + r + 8 * kh][ln] = acc[r];
      } else {
#pragma unroll
        for (int r = 0; r < 8; ++r) acc[r] = s_lin[r + 8 * kh][ln];
#pragma unroll
        for (int k = 0; k < 32; k += 4) {
          const int kk = k + 2 * kh;
          v2f a, b;
          a.x = Kc[kk][ln] * 0.5f; a.y = Kc[kk + 1][ln] * 0.5f;
          b.x = Vc[kk][ln];        b.y = Vc[kk + 1][ln];
          acc = WMMA_F32(a, b, acc);
        }
#pragma unroll
        for (int r = 0; r < 8; ++r) s_lin[r + 8 * kh][ln] = acc[r];
      }
    }
    __syncthreads();  // (E) end of chunk
  }
}

// ---------------------------------------------------------------------------
extern "C" void kernel_launch(void* const* d_in, const int* in_sizes, int n_in,
                              void* d_out, int out_size, void* d_ws, size_t ws_size,
                              hipStream_t stream) {
  (void)in_sizes; (void)n_in; (void)out_size; (void)ws_size;
  const float* hs = (const float*)d_in[0];
  const float* Wq = (const float*)d_in[1];
  const float* Wk = (const float*)d_in[2];
  const float* Wv = (const float*)d_in[3];
  const float* Wo = (const float*)d_in[4];
  float* out = (float*)d_out;

  float* Qg = (float*)d_ws;                  // 1024 x 192
  float* Kg = Qg + (size_t)L * (Hh * FD);    // 1024 x 192
  float* Vg = Kg + (size_t)L * (Hh * FD);    // 1024 x 768
  float* Yg = Vg + (size_t)L * DM;           // 1024 x 768

  const dim3 blk(256);
  gemm_f32_wmma<<<dim3((L / 64) * ((Hh * FD) / 64)), blk, 0, stream>>>(hs, Wq, Qg, L, Hh * FD, DM);
  gemm_f32_wmma<<<dim3((L / 64) * ((Hh * FD) / 64)), blk, 0, stream>>>(hs, Wk, Kg, L, Hh * FD, DM);
  gemm_f32_wmma<<<dim3((L / 64) * (DM / 64)), blk, 0, stream>>>(hs, Wv, Vg, L, DM, DM);
  based_attn<<<dim3(Hh * 4), blk, 0, stream>>>(Qg, Kg, Vg, Yg);
  gemm_f32_wmma<<<dim3((L / 64) * (DM / 64)), blk, 0, stream>>>(Yg, Wo, out, L, DM, DM);
}